// BilinearInterpolation_32856499814935
// MI455X (gfx1250) — compile-verified
//
#include <hip/hip_runtime.h>
#include <hip/hip_bf16.h>
#include <stdint.h>

typedef float v2f __attribute__((ext_vector_type(2)));
typedef float v8f __attribute__((ext_vector_type(8)));

namespace {
constexpr int kB  = 8;
constexpr int kC  = 3;
constexpr int kH  = 1024;
constexpr int kW  = 1024;
constexpr int kHW = kH * kW;
constexpr int kPixels = kB * kHW;   // 8388608 = 32768 * 256, exact grid
}

__global__ __launch_bounds__(256)
void warp_bilinear_wmma(const float* __restrict__ img,
                        const float* __restrict__ T,
                        float* __restrict__ out)
{
  const int  g        = blockIdx.x * 256 + threadIdx.x;  // this lane's pixel
  const int  lane     = threadIdx.x & 31;
  const bool lo       = (lane & 16) == 0;
  const int  waveBase = g & ~31;

  // ---- A matrix (16x4 f32): T^T placed in rows 0..2 AND 8..10 ------------
  // ISA layout: lane m (0-15) holds A[m][0] in VGPR0, A[m][1] in VGPR1;
  //             lane 16+m holds A[m][2] / A[m][3].
  const int  m     = lane & 15;
  const bool valid = (m < 3) || (m >= 8 && m < 11);
  const int  j     = (m < 3) ? m : (m - 8);
  float a0 = 0.0f, a1 = 0.0f;
  if (valid) {
    a0 = lo ? T[0 * 3 + j] : T[2 * 3 + j];   // A[m][0]=T[0][j], A[m][2]=T[2][j]
    a1 = lo ? T[1 * 3 + j] : 0.0f;           // A[m][1]=T[1][j], A[m][3]=0
  }
  v2f Amat; Amat[0] = a0; Amat[1] = a1;

  // ---- normalized grid coords (grid is x-major: xi=q/H, yi=q%H) ----------
  auto gx = [](int p) -> float {
    int xi = (p & (kHW - 1)) >> 10;
    return -1.0f + (float)xi * (2.0f / (float)(kW - 1));
  };
  auto gy = [](int p) -> float {
    int yi = p & (kH - 1);
    return -1.0f + (float)yi * (2.0f / (float)(kH - 1));
  };

  // ---- B matrix (4x16 f32): VGPR0 = {lo: K=0 (x), hi: K=2 (1.0)}
  //                           VGPR1 = {lo: K=1 (y), hi: K=3 (0.0)}
  const int pLo = waveBase + m;        // pixels 0..15 of the wave's tile
  const int pHi = waveBase + 16 + m;   // pixels 16..31 of the wave's tile

  v2f Blo; Blo[0] = lo ? gx(pLo) : 1.0f;  Blo[1] = lo ? gy(pLo) : 0.0f;
  v2f Bhi; Bhi[0] = lo ? gx(pHi) : 1.0f;  Bhi[1] = lo ? gy(pHi) : 0.0f;

  v8f c0 = {};
  v8f c1 = {};
  // D[m][n] = sum_k A[m][k]*B[k][n]; rows 0..2 -> lanes 0-15 VGPR 0..2,
  // duplicated rows 8..10 -> lanes 16-31 VGPR 0..2.
  c0 = __builtin_amdgcn_wmma_f32_16x16x4_f32(false, Amat, false, Blo,
                                             (short)0, c0, false, false);
  c1 = __builtin_amdgcn_wmma_f32_16x16x4_f32(false, Amat, false, Bhi,
                                             (short)0, c1, false, false);

  // lane n(<16) owns pixel waveBase+n (c0); lane 16+n owns waveBase+16+n (c1)
  const float s0 = lo ? c0[0] : c1[0];
  const float s1 = lo ? c0[1] : c1[1];
  const float s2 = lo ? c0[2] : c1[2];

  // ---- perspective divide (fast v_rcp_f32, <=1 ulp) + bilinear sample ----
  const float winv = __builtin_amdgcn_rcpf(s2);
  const float xf = 0.5f * (s0 * winv + 1.0f) * (float)kW;
  const float yf = 0.5f * (s1 * winv + 1.0f) * (float)kH;

  const int x0 = (int)rintf(xf);     // round half-to-even == jnp.round
  const int y0 = (int)rintf(yf);
  const int x0c = min(max(x0, 0), kW - 1);
  const int x1c = min(max(x0 + 1, 0), kW - 1);
  const int y0c = min(max(y0, 0), kH - 1);
  const int y1c = min(max(y0 + 1, 0), kH - 1);

  const int base = (g >> 20) * kHW;
  const int ia = (base + y0c * kW + x0c) * 3;   // reshape(-1,C) interleave
  const int ib = (base + y1c * kW + x0c) * 3;
  const int ic = (base + y0c * kW + x1c) * 3;
  const int id = (base + y1c * kW + x1c) * 3;

  auto q8 = [](float v) -> float { return (float)(uint8_t)(int)v; };

  float pa[3], pb[3], pc[3], pd[3];
#pragma unroll
  for (int c = 0; c < 3; ++c) {
    pa[c] = q8(img[ia + c]);
    pb[c] = q8(img[ib + c]);
    pc[c] = q8(img[ic + c]);
    pd[c] = q8(img[id + c]);
  }

  const float x0f = (float)x0c, x1f = (float)x1c;
  const float y0f = (float)y0c, y1f = (float)y1c;
  const float aa = (x1f - xf) * (y1f - yf);
  const float ab = (x1f - xf) * (yf - y0f);
  const float ac = (xf - x0f) * (y1f - yf);
  const float ad = (xf - x0f) * (yf - y0f);

  const int o = g * 3;
#pragma unroll
  for (int c = 0; c < 3; ++c) {
    out[o + c] = aa * pa[c] + ab * pb[c] + ac * pc[c] + ad * pd[c];
  }
}

extern "C" void kernel_launch(void* const* d_in, const int* in_sizes, int n_in,
                              void* d_out, int out_size, void* d_ws, size_t ws_size,
                              hipStream_t stream) {
  (void)in_sizes; (void)n_in; (void)out_size; (void)d_ws; (void)ws_size;
  const float* img = (const float*)d_in[0];
  const float* T   = (const float*)d_in[1];
  float*       out = (float*)d_out;
  const int blocks = kPixels / 256;   // 32768
  hipLaunchKernelGGL(warp_bilinear_wmma, dim3(blocks), dim3(256), 0, stream,
                     img, T, out);
}